// AdaptivePooling_69028714381469
// MI455X (gfx1250) — compile-verified
//
#include <hip/hip_runtime.h>
#include <hip/hip_bf16.h>

// AdaptivePooling (7x7 adaptive mean pool), NHWC fp32 (16,250,250,256) -> (16,7,7,256).
// HBM-bound: ~1.02 GB read @ 23.3 TB/s => ~44us floor; 0.25 FLOP/byte, so WMMA is
// irrelevant — the CDNA5 win is the Tensor Data Mover. Each (w,c) row-chunk of a bin is
// contiguous in NHWC, so one TDM descriptor streams ~37KB into LDS. This version
// double-buffers the TDM: issue row i+1, s_wait_tensorcnt(1) to prove row i landed
// (TDM ops complete in order per wave), consume row i from LDS while i+1 streams.

typedef __attribute__((ext_vector_type(4))) unsigned int v4u;
typedef __attribute__((ext_vector_type(8))) int          v8i;
typedef __attribute__((ext_vector_type(4))) int          v4i;

#if __has_builtin(__builtin_amdgcn_tensor_load_to_lds) && __has_builtin(__builtin_amdgcn_s_wait_tensorcnt)
#define POOL_HAS_TDM 1
#else
#define POOL_HAS_TDM 0
#endif

#define OUT_HW 7
#define IN_HW  250
#define CH     256
#define MAX_BINW 37

#if POOL_HAS_TDM
// Issue one TDM DMA: wc contiguous fp32 elements from `src` into LDS byte offset lds_off.
__device__ __forceinline__ void tdm_load_row(const float* src, unsigned lds_off, int wc) {
  unsigned long long ga = (unsigned long long)(const void*)src;
  v4u g0; v8i g1; v4i g2; v4i g3;
  // ---- D# group 0: count=1 | lds_addr | global_addr[56:0] | type=2 ----
  g0[0] = 1u;                                    // count=1 (valid), user mode
  g0[1] = lds_off;                               // lds_addr (bytes)
  g0[2] = (unsigned)(ga & 0xffffffffull);        // global_addr[31:0]
  g0[3] = ((unsigned)(ga >> 32) & 0x01ffffffu)   // global_addr[56:32]
          | 0x80000000u;                         // type=2 ("image") at [127:126]
  // ---- D# group 1: data_size=4B, 1-D tile of wc elements ----
  g1[0] = 0x00020000;                     // workgroup_mask=0, data_size=2 (4B), no flags
  g1[1] = (int)((wc & 0xffff) << 16);     // [47:32]=atomic_addr=0, [63:48]=tensor_dim0 lo16
  g1[2] = (int)((wc >> 16) & 0xffff)      // [79:64]=tensor_dim0 hi16
          | (1 << 16);                    // [95:80]=tensor_dim1 lo16 = 1
  g1[3] = (int)(wc << 16);                // [111:96]=tensor_dim1 hi16=0, [127:112]=tile_dim0
  g1[4] = 0;                              // tile_dim1=0 (unused), tile_dim2=0
  g1[5] = wc;                             // tensor_dim0_stride lo32
  g1[6] = (int)((IN_HW * CH & 0xffff) << 16);  // stride0 hi16=0, tensor_dim1_stride lo16
  g1[7] = (int)((IN_HW * CH) >> 16);           // tensor_dim1_stride [47:16]
  g2[0] = 0; g2[1] = 0; g2[2] = 0; g2[3] = 0;  // 2-D tensor: groups 2/3 unused
  g3[0] = 0; g3[1] = 0; g3[2] = 0; g3[3] = 0;
#if defined(__clang_major__) && (__clang_major__ >= 23)
  v8i g4; g4[0]=0; g4[1]=0; g4[2]=0; g4[3]=0; g4[4]=0; g4[5]=0; g4[6]=0; g4[7]=0;
  __builtin_amdgcn_tensor_load_to_lds(g0, g1, g2, g3, g4, 0);
#else
  __builtin_amdgcn_tensor_load_to_lds(g0, g1, g2, g3, 0);
#endif
}
#endif

__global__ __launch_bounds__(256) void AdaptivePooling_69028714381469_kernel(
    const float* __restrict__ x, float* __restrict__ out) {
  // Double-buffered (w,c) row chunks: 2 x 37 x 256 floats (~74KB) + reduction scratch.
  __shared__ __align__(16) float buf[2][MAX_BINW * CH];
  __shared__ float4 red[4][64];

  const int bid = blockIdx.x;
  const int b   = bid / (OUT_HW * OUT_HW);
  const int rr_ = bid % (OUT_HW * OUT_HW);
  const int oh  = rr_ / OUT_HW;
  const int ow  = rr_ % OUT_HW;

  // Bin edges exactly as the reference computes them (fp32 truncation / ceil, clamp).
  const float scale = 250.0f / 7.0f;
  int sh = (int)((float)oh * scale);
  int eh = (int)ceilf((float)(oh + 1) * scale); if (eh > IN_HW) eh = IN_HW;
  int sw = (int)((float)ow * scale);
  int ew = (int)ceilf((float)(ow + 1) * scale); if (ew > IN_HW) ew = IN_HW;
  const int wcount = ew - sw;     // 36 or 37
  const int rows   = eh - sh;     // 36 or 37
  const int wc     = wcount << 8; // fp32 elements in one contiguous row chunk

  const int t    = threadIdx.x;
  const int lane = t & 63;        // channel quad: channels [4*lane, 4*lane+3]
  const int grp  = t >> 6;        // w-group 0..3

  const float* rowp = x + ((((size_t)b * IN_HW + sh) * IN_HW + sw) << 8);
  const size_t rowstride = (size_t)IN_HW << 8;   // fp32 elements per h step

  float4 acc; acc.x = 0.f; acc.y = 0.f; acc.z = 0.f; acc.w = 0.f;

#if POOL_HAS_TDM
  const unsigned lds0 = (unsigned)(unsigned long long)(const void*)&buf[0][0];
  const unsigned lds1 = (unsigned)(unsigned long long)(const void*)&buf[1][0];
  if (t < 32) tdm_load_row(rowp, lds0, wc);   // prologue: row 0 -> buf[0]
#endif

  for (int i = 0; i < rows; ++i) {
    const int cur = i & 1;
#if POOL_HAS_TDM
    if (t < 32) {   // wave 0 drives the TDM; EXEC is ignored by tensor ops
      if (i + 1 < rows) {
        // Next row into the other buffer (that buffer was drained at end of iter i-1,
        // protected by the trailing __syncthreads()).
        tdm_load_row(rowp + (size_t)(i + 1) * rowstride, cur ? lds0 : lds1, wc);
        // TDM ops from one wave complete in order: <=1 outstanding => row i is in LDS.
        __builtin_amdgcn_s_wait_tensorcnt(1);
      } else {
        __builtin_amdgcn_s_wait_tensorcnt(0);
      }
    }
    __syncthreads();             // buf[cur] visible to all 8 waves
#else
    {  // Fallback: cooperative float4 copy (global_load_b128 -> ds_store_b128).
      const float* src = rowp + (size_t)i * rowstride;
      for (int j = t; j < (wc >> 2); j += 256)
        ((float4*)buf[cur])[j] = ((const float4*)src)[j];
      __syncthreads();
    }
#endif
    // Accumulate this row's bin columns: thread grp handles w = grp, grp+4, ...
    const float4* b4 = (const float4*)buf[cur];
    for (int wi = grp; wi < wcount; wi += 4) {
      float4 v = b4[(wi << 6) + lane];   // stride-64 float4: conflict-free banks
      acc.x += v.x; acc.y += v.y; acc.z += v.z; acc.w += v.w;
    }
    __syncthreads();             // drained: safe to re-target this buffer at iter i+1
  }

  // Reduce the 4 w-groups and write the 256-channel output pixel (global_store_b128).
  red[grp][lane] = acc;
  __syncthreads();
  if (grp == 0) {
    float4 a0 = red[0][lane], a1 = red[1][lane], a2 = red[2][lane], a3 = red[3][lane];
    const float inv = 1.0f / (float)(rows * wcount);
    float4 o;
    o.x = (a0.x + a1.x + a2.x + a3.x) * inv;
    o.y = (a0.y + a1.y + a2.y + a3.y) * inv;
    o.z = (a0.z + a1.z + a2.z + a3.z) * inv;
    o.w = (a0.w + a1.w + a2.w + a3.w) * inv;
    float4* op = (float4*)(out + ((((size_t)b * OUT_HW + oh) * OUT_HW + ow) << 8));
    op[lane] = o;
  }
}

extern "C" void kernel_launch(void* const* d_in, const int* in_sizes, int n_in,
                              void* d_out, int out_size, void* d_ws, size_t ws_size,
                              hipStream_t stream) {
  (void)in_sizes; (void)n_in; (void)out_size; (void)d_ws; (void)ws_size;
  const float* x = (const float*)d_in[0];
  float* out = (float*)d_out;
  const int nblocks = 16 * OUT_HW * OUT_HW;   // one block per (b, oh, ow)
  AdaptivePooling_69028714381469_kernel<<<dim3(nblocks), dim3(256), 0, stream>>>(x, out);
}